// GCN_89077621719557
// MI455X (gfx1250) — compile-verified
//
#include <hip/hip_runtime.h>
#include <hip/hip_bf16.h>

// GCN layer for MI455X (gfx1250, wave32):
//   h   = X @ W^T                      (WMMA f32 16x16x4 GEMM)
//   agg = segment_sum(val * h[src])    (gather + global_atomic_add_f32)
//   out = selu(h*skip + agg + bias)    (fused epilogue)

typedef __attribute__((ext_vector_type(2))) float v2f;
typedef __attribute__((ext_vector_type(8))) float v8f;

#define IN_CH  128
#define OUT_CH 64

// ---------------------------------------------------------------------------
// Kernel 1: h = X @ W^T via V_WMMA_F32_16X16X4_F32.
// Block = 256 threads = 8 waves. Each wave computes a 16(M) x 64(N) stripe:
// 4 accumulator tiles of 16x16, K=128 in steps of 4 -> 128 WMMAs per wave.
// W (64x128 f32 = 32 KB) is staged into LDS once per block.
// ---------------------------------------------------------------------------
__global__ __launch_bounds__(256) void gcn_gemm_wmma(
    const float* __restrict__ X,      // [nNodes, 128]
    const float* __restrict__ W,      // [64, 128]
    float* __restrict__ H,            // [nNodes, 64]
    int nNodes) {
  __shared__ float Wlds[OUT_CH * IN_CH];  // 32 KB

  const int tid = threadIdx.x;
  // cooperative load of W into LDS: 2048 float4 / 256 threads = 8 each
  {
    const float4* Wv = (const float4*)W;
    float4* Wl = (float4*)Wlds;
#pragma unroll
    for (int i = 0; i < (OUT_CH * IN_CH / 4) / 256; ++i)
      Wl[tid + i * 256] = Wv[tid + i * 256];
  }
  __syncthreads();

  const int wave = tid >> 5;        // 0..7
  const int lane = tid & 31;
  const int m0   = blockIdx.x * 128 + wave * 16;   // first row of this wave's tile
  const int ncol = lane & 15;                      // N within 16-wide tile
  const int khalf = (lane >> 4) * 2;               // lanes 0-15: K0/K1, 16-31: K2/K3

  // Clamp OOB rows to row 0 so EXEC stays all-1s through the WMMA loop
  // (WMMA requires EXEC == all ones); stores are predicated afterwards.
  int mrow = m0 + (lane & 15);
  if (mrow >= nNodes) mrow = 0;
  const float* __restrict__ Arow = X + (size_t)mrow * IN_CH + khalf;

  v8f acc0 = {}, acc1 = {}, acc2 = {}, acc3 = {};

  for (int k0 = 0; k0 < IN_CH; k0 += 4) {
    // A fragment 16x4: lane holds X[mrow][k0+khalf], X[mrow][k0+khalf+1]
    v2f a = *(const v2f*)(Arow + k0);
    // B fragment 4x16: B[k][n] = W[n][k]; lane supplies rows K=khalf,khalf+1 for col ncol
    v2f b0 = *(const v2f*)(&Wlds[(0  + ncol) * IN_CH + k0 + khalf]);
    v2f b1 = *(const v2f*)(&Wlds[(16 + ncol) * IN_CH + k0 + khalf]);
    v2f b2 = *(const v2f*)(&Wlds[(32 + ncol) * IN_CH + k0 + khalf]);
    v2f b3 = *(const v2f*)(&Wlds[(48 + ncol) * IN_CH + k0 + khalf]);
    // 8-arg form: (neg_a, A, neg_b, B, c_mod, C, reuse_a, reuse_b)
    acc0 = __builtin_amdgcn_wmma_f32_16x16x4_f32(false, a, false, b0, (short)0, acc0, false, false);
    acc1 = __builtin_amdgcn_wmma_f32_16x16x4_f32(false, a, false, b1, (short)0, acc1, false, false);
    acc2 = __builtin_amdgcn_wmma_f32_16x16x4_f32(false, a, false, b2, (short)0, acc2, false, false);
    acc3 = __builtin_amdgcn_wmma_f32_16x16x4_f32(false, a, false, b3, (short)0, acc3, false, false);
  }

  // D layout (32-bit C/D 16x16): VGPR r -> M = r (lanes 0-15) or M = r+8 (lanes 16-31),
  // N = lane & 15.
  const int rbase = m0 + (lane >> 4) * 8;
#pragma unroll
  for (int r = 0; r < 8; ++r) {
    const int m = rbase + r;
    if (m < nNodes) {
      float* hp = H + (size_t)m * OUT_CH + ncol;
      hp[0]  = acc0[r];
      hp[16] = acc1[r];
      hp[32] = acc2[r];
      hp[48] = acc3[r];
    }
  }
}

// ---------------------------------------------------------------------------
// Kernel 2: edge gather + scatter-add.
// One wave32 per edge, 2 contiguous channels per lane: the 256 B row read of
// h[src] is one fully-coalesced wave transaction; scatter uses f32 atomics.
// Each wave walks 4 consecutive edges to amortize index loads.
// ---------------------------------------------------------------------------
#define EDGES_PER_WAVE 4

__global__ __launch_bounds__(256) void gcn_spmm_edges(
    const int*   __restrict__ esrc,
    const int*   __restrict__ edst,
    const float* __restrict__ eval_,
    const float* __restrict__ H,      // [nNodes, 64]
    float*       __restrict__ agg,    // [nNodes, 64], pre-zeroed
    int nEdges) {
  const int lane = threadIdx.x & 31;
  const long wave = (long)blockIdx.x * (blockDim.x >> 5) + (threadIdx.x >> 5);
  long e = wave * EDGES_PER_WAVE;
#pragma unroll
  for (int i = 0; i < EDGES_PER_WAVE; ++i, ++e) {
    if (e >= nEdges) return;
    const int   s = esrc[e];
    const int   d = edst[e];
    const float v = eval_[e];
    const float2 hv = *(const float2*)(H + (size_t)s * OUT_CH + lane * 2);
    float* ap = agg + (size_t)d * OUT_CH + lane * 2;
    atomicAdd(ap + 0, v * hv.x);
    atomicAdd(ap + 1, v * hv.y);
  }
}

// ---------------------------------------------------------------------------
// Kernel 3: out = selu(h * skip_weight + agg + bias)
// ---------------------------------------------------------------------------
__global__ __launch_bounds__(256) void gcn_finalize(
    const float* __restrict__ H,
    const float* __restrict__ agg,
    const float* __restrict__ skipw,  // [64]
    const float* __restrict__ bias,   // [64]
    float* __restrict__ out,
    long total) {
  const long i = (long)blockIdx.x * blockDim.x + threadIdx.x;
  if (i >= total) return;
  const int c = (int)(i & (OUT_CH - 1));
  const float x = H[i] * skipw[c] + agg[i] + bias[c];
  const float alpha = 1.6732632423543772f;
  const float scale = 1.0507009873554805f;
  out[i] = x > 0.f ? scale * x : scale * alpha * (__expf(x) - 1.f);
}

// ---------------------------------------------------------------------------
// Kernel 0: zero the agg accumulator (workspace is not re-poisoned between
// replays, and the atomics accumulate, so this must run every launch).
// ---------------------------------------------------------------------------
__global__ __launch_bounds__(256) void gcn_zero(float4* __restrict__ p, long n4) {
  const long i = (long)blockIdx.x * blockDim.x + threadIdx.x;
  if (i < n4) p[i] = make_float4(0.f, 0.f, 0.f, 0.f);
}

extern "C" void kernel_launch(void* const* d_in, const int* in_sizes, int n_in,
                              void* d_out, int out_size, void* d_ws, size_t ws_size,
                              hipStream_t stream) {
  const float* features = (const float*)d_in[0];  // [N, 128]
  const int*   esrc     = (const int*)d_in[1];    // [E]
  const int*   edst     = (const int*)d_in[2];    // [E]
  const float* eval_    = (const float*)d_in[3];  // [E]
  const float* weight   = (const float*)d_in[4];  // [64, 128]
  const float* skipw    = (const float*)d_in[5];  // [64]
  const float* bias     = (const float*)d_in[6];  // [64]
  float* out = (float*)d_out;

  const int nNodes = in_sizes[0] / IN_CH;   // 100000
  const int nEdges = in_sizes[1];           // 1600000

  // Workspace layout: H [nNodes*64] f32 | agg [nNodes*64] f32  (~51.2 MB)
  float* H   = (float*)d_ws;
  float* agg = H + (size_t)nNodes * OUT_CH;

  // 0) zero agg
  const long n4 = (long)nNodes * OUT_CH / 4;
  gcn_zero<<<(unsigned)((n4 + 255) / 256), 256, 0, stream>>>((float4*)agg, n4);

  // 1) h = X @ W^T   (8 waves/block, 128 rows/block)
  gcn_gemm_wmma<<<(unsigned)((nNodes + 127) / 128), 256, 0, stream>>>(
      features, weight, H, nNodes);

  // 2) edge gather/scatter (1 edge per wave, 4 edges walked per wave)
  const long waves  = ((long)nEdges + EDGES_PER_WAVE - 1) / EDGES_PER_WAVE;
  const long blocks = (waves + 7) / 8;  // 8 waves per 256-thread block
  gcn_spmm_edges<<<(unsigned)blocks, 256, 0, stream>>>(
      esrc, edst, eval_, H, agg, nEdges);

  // 3) epilogue
  const long total = (long)nNodes * OUT_CH;
  gcn_finalize<<<(unsigned)((total + 255) / 256), 256, 0, stream>>>(
      H, agg, skipw, bias, out, total);
}